// ChallengingGeometricLoss_7378753815129
// MI455X (gfx1250) — compile-verified
//
#include <hip/hip_runtime.h>
#include <math.h>

typedef __attribute__((ext_vector_type(2))) float v2f;
typedef __attribute__((ext_vector_type(8))) float v8f;

#define N_ROWS 8192
#define DIM    512
#define BT     128   // block tile edge (I and J)
#define KC     32    // K slab staged through LDS
#define XS     36    // LDS row stride in floats (144B: 16B-aligned rows, conflict-free frag reads)
#define NSLAB  (DIM / KC)   // 16 K slabs

// ---------------------------------------------------------------------------
// CDNA5 async global -> LDS DMA (ASYNCcnt), per CDNA5 ISA.
// vdst = wave-relative LDS byte offset, vaddr = 64-bit global address (GV mode).
// ---------------------------------------------------------------------------
__device__ __forceinline__ void async_ld_b128(unsigned lds_byte_off, const float* gptr) {
  asm volatile("global_load_async_to_lds_b128 %0, %1, off"
               :: "v"(lds_byte_off), "v"(gptr)
               : "memory");
}
__device__ __forceinline__ void wait_asynccnt_le8() {
  asm volatile("s_wait_asynccnt 0x8" ::: "memory");
}
__device__ __forceinline__ void wait_asynccnt_0() {
  asm volatile("s_wait_asynccnt 0x0" ::: "memory");
}

__device__ __forceinline__ unsigned lds_off(const float* p) {
  return (unsigned)(uintptr_t)p;   // flat LDS addr: low 32 bits = LDS byte offset
}

// ---------------------------------------------------------------------------
// Kernel 1: squared L2 norm of each row. One wave (32 lanes) per row.
// ---------------------------------------------------------------------------
__global__ __launch_bounds__(256) void norms_kernel(const float* __restrict__ x,
                                                    float* __restrict__ norms) {
  const int wave = threadIdx.x >> 5;
  const int lane = threadIdx.x & 31;
  const int row  = blockIdx.x * 8 + wave;
  const float4* xr = (const float4*)(x + (size_t)row * DIM);
  float s = 0.f;
#pragma unroll
  for (int j = 0; j < 4; ++j) {          // 4 * 32 float4 = 512 floats, coalesced
    float4 v = xr[lane + 32 * j];
    s += v.x * v.x + v.y * v.y + v.z * v.z + v.w * v.w;
  }
#pragma unroll
  for (int m = 16; m >= 1; m >>= 1) s += __shfl_xor(s, m, 32);
  if (lane == 0) norms[row] = s;
}

// ---------------------------------------------------------------------------
// Kernel 2: upper-triangular 128x128 Gram tiles via V_WMMA_F32_16X16X4_F32,
// double-buffered async-DMA K slabs, fused dist->exp->sum epilogue.
// 8 waves (wave32) in a 2x4 grid; each wave owns a 64x32 sub-tile
// (4x2 accumulators of 16x16).
// ---------------------------------------------------------------------------
__global__ __launch_bounds__(256) void pair_kernel(const float* __restrict__ x,
                                                   const float* __restrict__ norms,
                                                   float* __restrict__ acc) {
  __shared__ float ldsBuf[2][2][BT * XS];   // [ping-pong][I/J][tile]  = 72 KB
  __shared__ float wsum[8];

  const int it = blockIdx.y;
  const int jt = blockIdx.x;
  if (jt < it) return;                       // symmetry: upper triangle only
  const float weight = (it == jt) ? 1.0f : 2.0f;

  const int tid  = threadIdx.x;
  const int wave = tid >> 5;
  const int lane = tid & 31;
  const int wi   = wave >> 2;                // 0..1 : I offset 64*wi
  const int wj   = wave & 3;                 // 0..3 : J offset 32*wj
  const int half = lane >> 4;                // 0/1  (K pair select for A/B frags)
  const int q    = lane & 15;                // row-in-tile / N index

  const size_t iBase = (size_t)it * BT;
  const size_t jBase = (size_t)jt * BT;

  // This thread's 4 cooperative-load slots (128 rows x 8 float4 per buffer).
  const int r0  = tid >> 3;                  // rows r0, r0+32, r0+64, r0+96
  const int c4  = tid & 7;                   // float4 column 0..7

  v8f c[4][2];
#pragma unroll
  for (int mi = 0; mi < 4; ++mi)
#pragma unroll
    for (int nj = 0; nj < 2; ++nj)
      c[mi][nj] = (v8f){0.f, 0.f, 0.f, 0.f, 0.f, 0.f, 0.f, 0.f};

  // Issue slab 0 into buffer 0 (8 async instructions per wave).
#pragma unroll
  for (int i = 0; i < 4; ++i) {
    const int r = r0 + i * 32;
    async_ld_b128(lds_off(&ldsBuf[0][0][r * XS + c4 * 4]), x + (iBase + r) * DIM + c4 * 4);
    async_ld_b128(lds_off(&ldsBuf[0][1][r * XS + c4 * 4]), x + (jBase + r) * DIM + c4 * 4);
  }

  for (int kt = 0; kt < NSLAB; ++kt) {
    // Prefetch slab kt+1 into the other buffer while slab kt is consumed.
    if (kt + 1 < NSLAB) {
      const int kb = (kt + 1) * KC;
      float* bI = &ldsBuf[(kt + 1) & 1][0][0];
      float* bJ = &ldsBuf[(kt + 1) & 1][1][0];
#pragma unroll
      for (int i = 0; i < 4; ++i) {
        const int r = r0 + i * 32;
        async_ld_b128(lds_off(bI + r * XS + c4 * 4), x + (iBase + r) * DIM + kb + c4 * 4);
        async_ld_b128(lds_off(bJ + r * XS + c4 * 4), x + (jBase + r) * DIM + kb + c4 * 4);
      }
      wait_asynccnt_le8();    // in-order completion => slab kt has landed
    } else {
      wait_asynccnt_0();
    }
    __syncthreads();          // all waves' slab-kt DMA visible in LDS

    const float* bI = &ldsBuf[kt & 1][0][0];
    const float* bJ = &ldsBuf[kt & 1][1][0];
#pragma unroll
    for (int k = 0; k < KC; k += 4) {
      // A frag (16x4 f32): lanes 0-15 hold K={k,k+1}, lanes 16-31 K={k+2,k+3}
      v2f a[4], b[2];
#pragma unroll
      for (int mi = 0; mi < 4; ++mi)
        a[mi] = *(const v2f*)(bI + (wi * 64 + mi * 16 + q) * XS + k + 2 * half);
#pragma unroll
      for (int nj = 0; nj < 2; ++nj)
        b[nj] = *(const v2f*)(bJ + (wj * 32 + nj * 16 + q) * XS + k + 2 * half);
#pragma unroll
      for (int mi = 0; mi < 4; ++mi)
#pragma unroll
        for (int nj = 0; nj < 2; ++nj)
          c[mi][nj] = __builtin_amdgcn_wmma_f32_16x16x4_f32(
              false, a[mi], false, b[nj], (short)0, c[mi][nj], false, false);
    }
    __syncthreads();          // nobody still reads buf[kt&1] before slab kt+2 overwrites it
  }

  // Fused epilogue: dist = sqrt(max(0, ni + nj - 2*S)), sum of exp(-0.1*dist).
  // C layout: lane l, vgpr v -> M = v + 8*(l>>4), N = l&15.
  float nj_val[2];
#pragma unroll
  for (int nj = 0; nj < 2; ++nj)
    nj_val[nj] = norms[jBase + wj * 32 + nj * 16 + q];

  float s = 0.f;
#pragma unroll
  for (int mi = 0; mi < 4; ++mi) {
#pragma unroll
    for (int v = 0; v < 8; ++v) {
      const float ni = norms[iBase + wi * 64 + mi * 16 + v + 8 * half];
#pragma unroll
      for (int nj = 0; nj < 2; ++nj) {
        float sq = ni + nj_val[nj] - 2.0f * c[mi][nj][v];
        sq = fmaxf(sq, 0.0f);
        s += __expf(-0.1f * __builtin_sqrtf(sq));
      }
    }
  }
  s *= weight;

#pragma unroll
  for (int m = 16; m >= 1; m >>= 1) s += __shfl_xor(s, m, 32);
  if (lane == 0) wsum[wave] = s;
  __syncthreads();
  if (tid == 0) {
    float t = 0.f;
#pragma unroll
    for (int w = 0; w < 8; ++w) t += wsum[w];
    atomicAdd(acc, t);                       // global_atomic_add_f32
  }
}

// ---------------------------------------------------------------------------
__global__ void zero_acc(float* acc) { acc[0] = 0.f; }

__global__ void finalize(const float* __restrict__ acc, float* __restrict__ out) {
  const float mean = acc[0] / ((float)N_ROWS * (float)N_ROWS);
  const float loss = mean * 0.1f;
  out[0] = loss;          // geometric_loss
  out[1] = 0.5f * loss;   // total_geometric = LAMBDA_GEO * loss
}

// ---------------------------------------------------------------------------
extern "C" void kernel_launch(void* const* d_in, const int* in_sizes, int n_in,
                              void* d_out, int out_size, void* d_ws, size_t ws_size,
                              hipStream_t stream) {
  (void)in_sizes; (void)n_in; (void)out_size; (void)ws_size;
  const float* x = (const float*)d_in[0];       // (4,2048,512) f32 -> (8192,512)
  float* out   = (float*)d_out;                 // [loss, 0.5*loss]
  float* acc   = (float*)d_ws;                  // ws[0]   : f32 accumulator
  float* norms = (float*)d_ws + 64;             // ws[64..]: 8192 row sq-norms

  zero_acc<<<1, 1, 0, stream>>>(acc);
  norms_kernel<<<N_ROWS / 8, 256, 0, stream>>>(x, norms);
  dim3 grid(N_ROWS / BT, N_ROWS / BT);          // 64 x 64 tiles, lower tri early-exits
  pair_kernel<<<grid, 256, 0, stream>>>(x, norms, acc);
  finalize<<<1, 1, 0, stream>>>(acc, out);
}